// TriDRABlock_2138893714267
// MI455X (gfx1250) — compile-verified
//
#include <hip/hip_runtime.h>

typedef __attribute__((ext_vector_type(2))) float v2f;
typedef __attribute__((ext_vector_type(8))) float v8f;

#define DEVFN __device__ __forceinline__

constexpr int L_TOT = 24 * 24 * 24;   // 13824
constexpr int CDIM  = 96;
constexpr int CDI   = 192;
constexpr int CDS   = 16;
constexpr int CDTR  = 6;
constexpr int XDW   = CDTR + 2 * CDS; // 38
constexpr int LQ    = L_TOT / 4;      // 3456

DEVFN float siluf(float x)     { return x / (1.f + expf(-x)); }
DEVFN float sigmf(float x)     { return 1.f / (1.f + expf(-x)); }
DEVFN float softplusf(float x) { return (x > 20.f) ? x : log1pf(expf(x)); }
DEVFN float gelu_ex(float x)   { return 0.5f * x * (1.f + erff(x * 0.70710678118654752f)); }

// sequence index -> original token index for branch br
DEVFN int seq_map(int br, int t) {
  if (br == 0) return t;
  if (br == 1) return L_TOT - 1 - t;
  return (t & 3) * LQ + (t >> 2);   // PERM[t]
}

// ---------------------------------------------------------------------------
// LayerNorm over channels (x is channel-major (C,L); out token-major (L,C))
// ---------------------------------------------------------------------------
__global__ void ln_kernel(const float* __restrict__ x, const float* __restrict__ w,
                          const float* __restrict__ b, float* __restrict__ xn) {
  int l = blockIdx.x;
  int c = threadIdx.x;           // 0..127
  __shared__ float s1[128], s2[128];
  float v = (c < CDIM) ? x[(size_t)c * L_TOT + l] : 0.f;
  s1[c] = v; s2[c] = v * v;
  __syncthreads();
  for (int off = 64; off; off >>= 1) {
    if (c < off) { s1[c] += s1[c + off]; s2[c] += s2[c + off]; }
    __syncthreads();
  }
  float mu  = s1[0] / (float)CDIM;
  float var = s2[0] / (float)CDIM - mu * mu;
  float rs  = rsqrtf(var + 1e-5f);
  if (c < CDIM) xn[(size_t)l * CDIM + c] = (v - mu) * rs * w[c] + b[c];
}

// ---------------------------------------------------------------------------
// Generic f32 WMMA GEMM: C(MxN) = A(MxK) @ W(N,K)^T [+ bias]
// ep==0: plain store (ldc). ep==1: in_proj split: n<192 -> C0, else silu->C1
// One 16x16 tile per wave, 8 waves/block. Branch-free operand fetch:
// out-of-range rows/cols are CLAMPED (a D column depends only on its own B
// column, so garbage in clamped columns never reaches stored results).
// ---------------------------------------------------------------------------
__global__ void gemm_wmma(const float* __restrict__ A, const float* __restrict__ W,
                          const float* __restrict__ bias,
                          float* __restrict__ C0, float* __restrict__ C1,
                          int M, int N, int K, int ldc, int ep) {
  const int wave = threadIdx.x >> 5;
  const int lane = threadIdx.x & 31;
  const int m0 = (blockIdx.y * 2 + (wave >> 2)) * 16;
  const int n0 = (blockIdx.x * 4 + (wave & 3)) * 16;
  if (m0 >= M || n0 >= N) return;          // wave-uniform
  const int mrow_l = m0 + (lane & 15);
  const int ncol_l = n0 + (lane & 15);
  const int mrow = (mrow_l < M) ? mrow_l : (M - 1);
  const int ncol = (ncol_l < N) ? ncol_l : (N - 1);
  const int koff = (lane >> 4) << 1;       // 0 or 2
  const float* Arow = A + (size_t)mrow * K + koff;
  const float* Wrow = W + (size_t)ncol * K + koff;
  v8f acc = {};
#pragma unroll 4
  for (int k = 0; k < K; k += 4) {
    v2f a = *(const v2f*)(Arow + k);
    v2f b = *(const v2f*)(Wrow + k);
    acc = __builtin_amdgcn_wmma_f32_16x16x4_f32(false, a, false, b, (short)0, acc,
                                                false, false);
  }
  const int   mb  = m0 + ((lane >> 4) << 3);
  const bool  nok = (ncol_l < N);
  const float bv  = (bias != nullptr) ? bias[ncol] : 0.f;
#pragma unroll
  for (int r = 0; r < 8; ++r) {
    int m = mb + r;
    if (m < M && nok) {
      float v = acc[r] + bv;
      if (ep == 0) {
        C0[(size_t)m * ldc + ncol_l] = v;
      } else {                              // in_proj split-silu
        if (ncol_l < CDI) C0[(size_t)m * CDI + ncol_l] = v;
        else              C1[(size_t)m * CDI + (ncol_l - CDI)] = siluf(v);
      }
    }
  }
}

// ---------------------------------------------------------------------------
// repack fus conv weight (co,ci,tap) -> (tap,co,ci) so B fragments are
// contiguous v2f loads in the implicit-GEMM conv
// ---------------------------------------------------------------------------
__global__ void repack_w(const float* __restrict__ src, float* __restrict__ dst) {
  int idx = blockIdx.x * blockDim.x + threadIdx.x;
  if (idx >= CDIM * CDI * 27) return;
  int co = idx / (CDI * 27), rr = idx % (CDI * 27);
  int ci = rr / 27, tap = rr % 27;
  dst[((size_t)tap * CDIM + co) * CDI + ci] = src[idx];
}

// ---------------------------------------------------------------------------
// Implicit-GEMM dense 3x3x3 conv: out(l,co) = sum_tap sum_ci cat(l',ci)*W
// cat = [A0 (96ch) | A1 (96ch)], Wt repacked (tap,co,ci). Halo handled by
// clamping the neighbor row + multiplying the A fragment by a 0/1 mask.
// ---------------------------------------------------------------------------
__global__ void conv3_gemm(const float* __restrict__ A0, const float* __restrict__ A1,
                           const float* __restrict__ Wt, const float* __restrict__ bias,
                           float* __restrict__ Cout) {
  const int wave = threadIdx.x >> 5;
  const int lane = threadIdx.x & 31;
  const int m0 = (blockIdx.y * 2 + (wave >> 2)) * 16;
  const int n0 = (blockIdx.x * 4 + (wave & 3)) * 16;
  if (m0 >= L_TOT || n0 >= CDIM) return;   // wave-uniform
  const int mrow = m0 + (lane & 15);
  const int ncol = n0 + (lane & 15);
  const int koff = (lane >> 4) << 1;
  const int h = mrow / 576, rem = mrow % 576, w = rem / 24, z = rem % 24;
  v8f acc = {};
  for (int tap = 0; tap < 27; ++tap) {
    const int hh = h + tap / 9 - 1;
    const int ww = w + (tap / 3) % 3 - 1;
    const int zz = z + tap % 3 - 1;
    const bool valid = (hh >= 0 && hh < 24 && ww >= 0 && ww < 24 && zz >= 0 && zz < 24);
    const float msk = valid ? 1.f : 0.f;
    const int lp = valid ? ((hh * 24 + ww) * 24 + zz) : mrow;  // clamp to own row
    const float* r0 = A0 + (size_t)lp * CDIM + koff;
    const float* r1 = A1 + (size_t)lp * CDIM + koff;
    const float* w0 = Wt + ((size_t)tap * CDIM + ncol) * CDI + koff;
#pragma unroll 4
    for (int k = 0; k < CDIM; k += 4) {          // ci = 0..95 (out_m half)
      v2f a = *(const v2f*)(r0 + k);
      a.x *= msk; a.y *= msk;
      v2f b = *(const v2f*)(w0 + k);
      acc = __builtin_amdgcn_wmma_f32_16x16x4_f32(false, a, false, b, (short)0, acc,
                                                  false, false);
    }
#pragma unroll 4
    for (int k = 0; k < CDIM; k += 4) {          // ci = 96..191 (refined half)
      v2f a = *(const v2f*)(r1 + k);
      a.x *= msk; a.y *= msk;
      v2f b = *(const v2f*)(w0 + CDIM + k);
      acc = __builtin_amdgcn_wmma_f32_16x16x4_f32(false, a, false, b, (short)0, acc,
                                                  false, false);
    }
  }
  const int mb = m0 + ((lane >> 4) << 3);
  const float bv = bias[ncol];
#pragma unroll
  for (int r = 0; r < 8; ++r)
    Cout[(size_t)(mb + r) * CDIM + ncol] = acc[r] + bv;
}

// ---------------------------------------------------------------------------
// causal depthwise conv1d (k=4) + silu for all 3 branches (index-mapped input)
// ---------------------------------------------------------------------------
__global__ void conv1d_silu(const float* __restrict__ u, const float* __restrict__ cw,
                            const float* __restrict__ cb, float* __restrict__ uc) {
  long long idx = (long long)blockIdx.x * blockDim.x + threadIdx.x;
  if (idx >= (long long)3 * L_TOT * CDI) return;
  int br = (int)(idx / ((long long)L_TOT * CDI));
  int rr = (int)(idx % ((long long)L_TOT * CDI));
  int t = rr / CDI, d = rr % CDI;
  float s = cb[br * CDI + d];
  const float* wk = cw + ((size_t)br * CDI + d) * 4;
#pragma unroll
  for (int j = 0; j < 4; ++j) {
    int tt = t - 3 + j;
    if (tt >= 0) s += wk[j] * u[(size_t)seq_map(br, tt) * CDI + d];
  }
  uc[idx] = siluf(s);
}

// dt = softplus(xd[:, :6] @ dtw^T + dtb)
__global__ void dt_kernel(const float* __restrict__ xd, const float* __restrict__ dtw,
                          const float* __restrict__ dtb, float* __restrict__ dt) {
  long long idx = (long long)blockIdx.x * blockDim.x + threadIdx.x;
  if (idx >= (long long)3 * L_TOT * CDI) return;
  int br = (int)(idx / ((long long)L_TOT * CDI));
  int rr = (int)(idx % ((long long)L_TOT * CDI));
  int t = rr / CDI, d = rr % CDI;
  const float* xrow = xd + ((size_t)br * L_TOT + t) * XDW;
  const float* wk = dtw + ((size_t)br * CDI + d) * CDTR;
  float s = dtb[br * CDI + d];
#pragma unroll
  for (int r = 0; r < CDTR; ++r) s += xrow[r] * wk[r];
  dt[idx] = softplusf(s);
}

// ---------------------------------------------------------------------------
// selective scan: 16 lanes per channel (one per state), shfl reduction,
// result gated by sz and scattered back to original token order
// ---------------------------------------------------------------------------
__global__ void scan_kernel(const float* __restrict__ uc, const float* __restrict__ dt,
                            const float* __restrict__ xd, const float* __restrict__ A_log,
                            const float* __restrict__ Dv, const float* __restrict__ sz,
                            float* __restrict__ ysg) {
  const int br = blockIdx.y;
  const int n = threadIdx.x & 15;
  const int d = blockIdx.x * (blockDim.x >> 4) + (threadIdx.x >> 4);
  const float* ucb = uc + (size_t)br * L_TOT * CDI;
  const float* dtb = dt + (size_t)br * L_TOT * CDI;
  const float* xdb = xd + (size_t)br * L_TOT * XDW;
  float* yb = ysg + (size_t)br * L_TOT * CDI;
  const float A  = -expf(A_log[((size_t)br * CDI + d) * CDS + n]);
  const float Dd = Dv[br * CDI + d];
  float h = 0.f;
  for (int t = 0; t < L_TOT; ++t) {
    const float dtv = dtb[(size_t)t * CDI + d];
    const float uv  = ucb[(size_t)t * CDI + d];
    const float Bv  = xdb[(size_t)t * XDW + CDTR + n];
    const float Cv  = xdb[(size_t)t * XDW + CDTR + CDS + n];
    h = h * expf(dtv * A) + dtv * uv * Bv;
    float p = h * Cv;
    p += __shfl_xor(p, 1, 16);
    p += __shfl_xor(p, 2, 16);
    p += __shfl_xor(p, 4, 16);
    p += __shfl_xor(p, 8, 16);
    if (n == 0) {
      const int po = seq_map(br, t);
      yb[(size_t)po * CDI + d] = (p + uv * Dd) * sz[(size_t)po * CDI + d];
    }
  }
}

// ---------------------------------------------------------------------------
// small elementwise / reduction kernels
// ---------------------------------------------------------------------------
__global__ void sum3_kernel(const float* __restrict__ y, float* __restrict__ s) {
  long long idx = (long long)blockIdx.x * blockDim.x + threadIdx.x;
  if (idx >= (long long)L_TOT * CDI) return;
  const size_t st = (size_t)L_TOT * CDI;
  s[idx] = y[idx] + y[idx + st] + y[idx + 2 * st];
}

__global__ void tok_means(const float* __restrict__ mbase, float* __restrict__ toks) {
  int c = blockIdx.x, i = blockIdx.y, tid = threadIdx.x;
  const float* m = mbase + (size_t)i * L_TOT * CDIM;
  __shared__ float s[256];
  float a = 0.f;
  for (int l = tid; l < L_TOT; l += 256) a += m[(size_t)l * CDIM + c];
  s[tid] = a; __syncthreads();
  for (int off = 128; off; off >>= 1) {
    if (tid < off) s[tid] += s[tid + off];
    __syncthreads();
  }
  if (tid == 0) toks[i * CDIM + c] = s[0] / (float)L_TOT;
}

__global__ void route_kernel(const float* __restrict__ toks,
                             const float* __restrict__ w1, const float* __restrict__ b1,
                             const float* __restrict__ w2, const float* __restrict__ b2,
                             float* __restrict__ route) {
  __shared__ float tn[3 * CDIM];
  __shared__ float rel[9];
  __shared__ float hid[48];
  int tid = threadIdx.x;                 // blockDim = 96
  tn[tid] = toks[tid]; tn[96 + tid] = toks[96 + tid]; tn[192 + tid] = toks[192 + tid];
  __syncthreads();
  if (tid < 3) {
    float s = 0.f;
    for (int d2 = 0; d2 < CDIM; ++d2) { float v = tn[tid * CDIM + d2]; s += v * v; }
    float nrm = fmaxf(sqrtf(s), 1e-12f);
    for (int d2 = 0; d2 < CDIM; ++d2) tn[tid * CDIM + d2] /= nrm;
  }
  __syncthreads();
  if (tid < 9) {
    int i = tid / 3, j = tid % 3;
    float s = 0.f;
    for (int d2 = 0; d2 < CDIM; ++d2) s += tn[i * CDIM + d2] * tn[j * CDIM + d2];
    rel[tid] = s;
  }
  __syncthreads();
  if (tid < 48) {
    float s = b1[tid];
    for (int k = 0; k < 9; ++k) s += rel[k] * w1[tid * 9 + k];
    hid[tid] = gelu_ex(s);
  }
  __syncthreads();
  if (tid == 0) {
    float lg[3], mx = -1e30f;
    for (int i = 0; i < 3; ++i) {
      float s = b2[i];
      for (int hh = 0; hh < 48; ++hh) s += hid[hh] * w2[i * 48 + hh];
      lg[i] = s; mx = fmaxf(mx, s);
    }
    float den = 0.f;
    for (int i = 0; i < 3; ++i) { lg[i] = expf(lg[i] - mx); den += lg[i]; }
    for (int i = 0; i < 3; ++i) route[i] = lg[i] / den;
  }
}

__global__ void fuse_comb(const float* __restrict__ m0, const float* __restrict__ m1,
                          const float* __restrict__ m2, const float* __restrict__ route,
                          float* __restrict__ fused) {
  long long idx = (long long)blockIdx.x * blockDim.x + threadIdx.x;
  if (idx >= (long long)L_TOT * CDIM) return;
  fused[idx] = route[0] * m0[idx] + route[1] * m1[idx] + route[2] * m2[idx];
}

__global__ void dwconv3d(const float* __restrict__ in, const float* __restrict__ wgt,
                         const float* __restrict__ bias, float* __restrict__ out) {
  long long idx = (long long)blockIdx.x * blockDim.x + threadIdx.x;
  if (idx >= (long long)L_TOT * CDIM) return;
  int c = (int)(idx % CDIM), l = (int)(idx / CDIM);
  int h = l / 576, rem = l % 576, w = rem / 24, z = rem % 24;
  float s = bias[c];
#pragma unroll
  for (int tap = 0; tap < 27; ++tap) {
    int hh = h + tap / 9 - 1, ww = w + (tap / 3) % 3 - 1, zz = z + tap % 3 - 1;
    if (hh < 0 || hh >= 24 || ww < 0 || ww >= 24 || zz < 0 || zz >= 24) continue;
    s += in[(size_t)((hh * 24 + ww) * 24 + zz) * CDIM + c] * wgt[c * 27 + tap];
  }
  out[idx] = s;
}

__global__ void chan_stats(const float* __restrict__ buf, float* __restrict__ mean,
                           float* __restrict__ rstd) {
  int c = blockIdx.x, tid = threadIdx.x;
  __shared__ float s1[256], s2[256];
  float a = 0.f, b = 0.f;
  for (int l = tid; l < L_TOT; l += 256) {
    float v = buf[(size_t)l * CDIM + c];
    a += v; b += v * v;
  }
  s1[tid] = a; s2[tid] = b; __syncthreads();
  for (int off = 128; off; off >>= 1) {
    if (tid < off) { s1[tid] += s1[tid + off]; s2[tid] += s2[tid + off]; }
    __syncthreads();
  }
  if (tid == 0) {
    float m = s1[0] / (float)L_TOT;
    float v = s2[0] / (float)L_TOT - m * m;
    mean[c] = m; rstd[c] = rsqrtf(v + 1e-5f);
  }
}

__global__ void inorm_gelu(float* __restrict__ buf, const float* __restrict__ mean,
                           const float* __restrict__ rstd) {
  long long idx = (long long)blockIdx.x * blockDim.x + threadIdx.x;
  if (idx >= (long long)L_TOT * CDIM) return;
  int c = (int)(idx % CDIM);
  buf[idx] = gelu_ex((buf[idx] - mean[c]) * rstd[c]);
}

__global__ void refine_gate(const float* __restrict__ fused, const float* __restrict__ lr,
                            float* __restrict__ refined) {
  long long idx = (long long)blockIdx.x * blockDim.x + threadIdx.x;
  if (idx >= (long long)L_TOT * CDIM) return;
  refined[idx] = fused[idx] * sigmf(lr[idx]);
}

__global__ void final_add(const float* __restrict__ fu, const float* __restrict__ x,
                          float* __restrict__ out) {
  long long idx = (long long)blockIdx.x * blockDim.x + threadIdx.x;
  if (idx >= (long long)L_TOT * CDIM) return;
  int c = (int)(idx % CDIM), l = (int)(idx / CDIM);
  out[(size_t)c * L_TOT + l] = fu[idx] + x[(size_t)c * L_TOT + l];
}

// ---------------------------------------------------------------------------
// host orchestration
// ---------------------------------------------------------------------------
extern "C" void kernel_launch(void* const* d_in, const int* in_sizes, int n_in,
                              void* d_out, int out_size, void* d_ws, size_t ws_size,
                              hipStream_t stream) {
  const float* x         = (const float*)d_in[0];
  const float* ln_w      = (const float*)d_in[1];
  const float* ln_b      = (const float*)d_in[2];
  const float* in_proj_w = (const float*)d_in[3];
  const float* conv_w    = (const float*)d_in[4];
  const float* conv_b    = (const float*)d_in[5];
  const float* xproj_w   = (const float*)d_in[6];
  const float* dtproj_w  = (const float*)d_in[7];
  const float* dtproj_b  = (const float*)d_in[8];
  const float* A_log     = (const float*)d_in[9];
  const float* Dv        = (const float*)d_in[10];
  const float* out_proj_w= (const float*)d_in[11];
  const float* dirproj_w = (const float*)d_in[12];
  const float* dirproj_b = (const float*)d_in[13];
  const float* rel_w1    = (const float*)d_in[14];
  const float* rel_b1    = (const float*)d_in[15];
  const float* rel_w2    = (const float*)d_in[16];
  const float* rel_b2    = (const float*)d_in[17];
  const float* lr_dw_w   = (const float*)d_in[18];
  const float* lr_dw_b   = (const float*)d_in[19];
  const float* lr_pw_w   = (const float*)d_in[20];
  const float* lr_pw_b   = (const float*)d_in[21];
  const float* fus_w1    = (const float*)d_in[22];
  const float* fus_b1    = (const float*)d_in[23];
  const float* fus_pw_w  = (const float*)d_in[24];
  const float* fus_pw_b  = (const float*)d_in[25];
  float* out = (float*)d_out;
  float* ws  = (float*)d_ws;

  // workspace layout (floats), with aliasing of dead regions
  const size_t OFF_U   = 0;                                   // L*192  u (later sum_s)
  const size_t OFF_SZ  = OFF_U  + (size_t)L_TOT * CDI;        // L*192  silu(zg)
  const size_t OFF_UC  = OFF_SZ + (size_t)L_TOT * CDI;        // 3*L*192 uc (later lr/fu chain)
  const size_t OFF_XD  = OFF_UC + (size_t)3 * L_TOT * CDI;    // 3*L*38 xd (later out_seq)
  const size_t OFF_DT  = OFF_XD + (size_t)3 * L_TOT * XDW;    // 3*L*192 dt (later m's + fused)
  const size_t OFF_YSG = OFF_DT + (size_t)3 * L_TOT * CDI;    // 3*L*192 gated ys (first: xn)
  const size_t OFF_SM  = OFF_YSG+ (size_t)3 * L_TOT * CDI;    // small scalars
  const size_t OFF_WT  = OFF_SM + 1024;                       // repacked fus weight
  const size_t OFF_XN    = OFF_YSG;                 // L*96, dead before scan
  const size_t OFF_SUMS  = OFF_U;                   // L*192, u dead after conv1d
  const size_t OFF_OSEQ  = OFF_XD;                  // L*96, xd dead after scan
  const size_t OFF_M     = OFF_DT;                  // 3*L*96, dt dead after scan
  const size_t OFF_FUSED = OFF_DT + (size_t)3 * L_TOT * CDIM;
  const size_t OFF_LR1   = OFF_UC;                  // uc dead after scan
  const size_t OFF_LR3   = OFF_UC + (size_t)1 * L_TOT * CDIM;
  const size_t OFF_REF   = OFF_UC + (size_t)2 * L_TOT * CDIM;
  const size_t OFF_FU1   = OFF_UC + (size_t)3 * L_TOT * CDIM;
  const size_t OFF_FU3   = OFF_UC + (size_t)4 * L_TOT * CDIM;
  const size_t OFF_TOKS  = OFF_SM;
  const size_t OFF_ROUTE = OFF_SM + 512;
  const size_t OFF_MEAN  = OFF_SM + 768;
  const size_t OFF_RSTD  = OFF_SM + 896;

  const dim3 blk256(256);
  const dim3 ew_grid((L_TOT * CDIM + 255) / 256);
  const dim3 big_grid((3LL * L_TOT * CDI + 255) / 256);

  // 0. repack fus conv weight (co,ci,tap)->(tap,co,ci)
  repack_w<<<dim3((CDIM * CDI * 27 + 255) / 256), blk256, 0, stream>>>(fus_w1, ws + OFF_WT);

  // 1. LayerNorm -> xn (L x 96)
  ln_kernel<<<dim3(L_TOT), dim3(128), 0, stream>>>(x, ln_w, ln_b, ws + OFF_XN);

  // 2. in_proj GEMM (L x 96) @ (96 x 384) split -> u, sz=silu(zg)
  gemm_wmma<<<dim3((2 * CDI + 63) / 64, (L_TOT + 31) / 32), blk256, 0, stream>>>(
      ws + OFF_XN, in_proj_w, nullptr, ws + OFF_U, ws + OFF_SZ,
      L_TOT, 2 * CDI, CDIM, 0, 1);

  // 3. causal depthwise conv1d + silu for 3 branches -> uc
  conv1d_silu<<<big_grid, blk256, 0, stream>>>(ws + OFF_U, conv_w, conv_b, ws + OFF_UC);

  // 4. x_proj GEMMs: (L x 192) @ (192 x 38) -> xd[br]
  for (int br = 0; br < 3; ++br)
    gemm_wmma<<<dim3((XDW + 63) / 64, (L_TOT + 31) / 32), blk256, 0, stream>>>(
        ws + OFF_UC + (size_t)br * L_TOT * CDI, xproj_w + (size_t)br * XDW * CDI, nullptr,
        ws + OFF_XD + (size_t)br * L_TOT * XDW, nullptr, L_TOT, XDW, CDI, XDW, 0);

  // 5. dt = softplus(xd[:, :6] @ dtw^T + dtb)
  dt_kernel<<<big_grid, blk256, 0, stream>>>(ws + OFF_XD, dtproj_w, dtproj_b, ws + OFF_DT);

  // 6. selective scans (gated by sz, scattered back to original order) -> ysg
  scan_kernel<<<dim3(CDI / 16, 3), blk256, 0, stream>>>(
      ws + OFF_UC, ws + OFF_DT, ws + OFF_XD, A_log, Dv, ws + OFF_SZ, ws + OFF_YSG);

  // 7. sum of 3 gated branches -> sum_s
  sum3_kernel<<<dim3(((size_t)L_TOT * CDI + 255) / 256), blk256, 0, stream>>>(
      ws + OFF_YSG, ws + OFF_SUMS);

  // 8. out_proj GEMM: (L x 192) @ (192 x 96) -> out_seq
  gemm_wmma<<<dim3(2, (L_TOT + 31) / 32), blk256, 0, stream>>>(
      ws + OFF_SUMS, out_proj_w, nullptr, ws + OFF_OSEQ, nullptr,
      L_TOT, CDIM, CDI, CDIM, 0);

  // 9. dirproj GEMMs -> fwd_m/bwd_m/slc_m
  for (int br = 0; br < 3; ++br)
    gemm_wmma<<<dim3(2, (L_TOT + 31) / 32), blk256, 0, stream>>>(
        ws + OFF_YSG + (size_t)br * L_TOT * CDI, dirproj_w, dirproj_b,
        ws + OFF_M + (size_t)br * L_TOT * CDIM, nullptr, L_TOT, CDIM, CDI, CDIM, 0);

  // 10. token means -> toks
  tok_means<<<dim3(CDIM, 3), blk256, 0, stream>>>(ws + OFF_M, ws + OFF_TOKS);

  // 11. routing MLP -> route[3]
  route_kernel<<<dim3(1), dim3(96), 0, stream>>>(
      ws + OFF_TOKS, rel_w1, rel_b1, rel_w2, rel_b2, ws + OFF_ROUTE);

  // 12. fused = r0*fwd_m + r1*bwd_m + r2*slc_m
  fuse_comb<<<ew_grid, blk256, 0, stream>>>(
      ws + OFF_M, ws + OFF_M + (size_t)L_TOT * CDIM, ws + OFF_M + (size_t)2 * L_TOT * CDIM,
      ws + OFF_ROUTE, ws + OFF_FUSED);

  // 13. depthwise 3x3x3 conv -> lr1
  dwconv3d<<<ew_grid, blk256, 0, stream>>>(ws + OFF_FUSED, lr_dw_w, lr_dw_b, ws + OFF_LR1);

  // 14-15. instance norm + gelu (in place)
  chan_stats<<<dim3(CDIM), blk256, 0, stream>>>(ws + OFF_LR1, ws + OFF_MEAN, ws + OFF_RSTD);
  inorm_gelu<<<ew_grid, blk256, 0, stream>>>(ws + OFF_LR1, ws + OFF_MEAN, ws + OFF_RSTD);

  // 16. lr_pw GEMM (L x 96) @ (96 x 96) + b -> lr3
  gemm_wmma<<<dim3(2, (L_TOT + 31) / 32), blk256, 0, stream>>>(
      ws + OFF_LR1, lr_pw_w, lr_pw_b, ws + OFF_LR3, nullptr, L_TOT, CDIM, CDIM, CDIM, 0);

  // 17. refined = fused * sigmoid(lr3)
  refine_gate<<<ew_grid, blk256, 0, stream>>>(ws + OFF_FUSED, ws + OFF_LR3, ws + OFF_REF);

  // 18. dense 3x3x3 conv over cat([out_m, refined]) via implicit WMMA GEMM -> fu1
  conv3_gemm<<<dim3(2, (L_TOT + 31) / 32), blk256, 0, stream>>>(
      ws + OFF_OSEQ, ws + OFF_REF, ws + OFF_WT, fus_b1, ws + OFF_FU1);

  // 19-20. instance norm + gelu (in place)
  chan_stats<<<dim3(CDIM), blk256, 0, stream>>>(ws + OFF_FU1, ws + OFF_MEAN, ws + OFF_RSTD);
  inorm_gelu<<<ew_grid, blk256, 0, stream>>>(ws + OFF_FU1, ws + OFF_MEAN, ws + OFF_RSTD);

  // 21. fus_pw GEMM + b -> fu3
  gemm_wmma<<<dim3(2, (L_TOT + 31) / 32), blk256, 0, stream>>>(
      ws + OFF_FU1, fus_pw_w, fus_pw_b, ws + OFF_FU3, nullptr, L_TOT, CDIM, CDIM, CDIM, 0);

  // 22. residual add, transpose to channel-major output
  final_add<<<ew_grid, blk256, 0, stream>>>(ws + OFF_FU3, x, out);
}